// GNNRegressionModel_13451837571700
// MI455X (gfx1250) — compile-verified
//
#include <hip/hip_runtime.h>
#include <hip/hip_bf16.h>

typedef __attribute__((ext_vector_type(2))) float v2f;
typedef __attribute__((ext_vector_type(8))) float v8f;

#define HDIM 128

// ---------------- degree / dinv ----------------
__global__ void k_deg_init(float* dinv, int n) {
  int i = blockIdx.x * blockDim.x + threadIdx.x;
  if (i < n) dinv[i] = 1.0f;  // self-loop counts as 1
}
__global__ void k_deg_count(const int* __restrict__ dst, float* dinv, int E) {
  int i = blockIdx.x * blockDim.x + threadIdx.x;
  int stride = gridDim.x * blockDim.x;
  for (int e = i; e < E; e += stride) atomicAdd(&dinv[dst[e]], 1.0f);
}
__global__ void k_deg_fin(float* dinv, int n) {
  int i = blockIdx.x * blockDim.x + threadIdx.x;
  if (i < n) dinv[i] = rsqrtf(dinv[i]);
}

// ---------------- WMMA GEMM: H = act(A) @ W,  optional AGG = H*dinv[row]^2 + bias[col]
// A: [n,128], W: [128,128] row-major (row stride 128). One block = 8 waves,
// wave w computes the 16x16 tile (rows blockIdx.x*16, cols w*16), K=128 via 32x wmma_f32_16x16x4_f32.
// RELU and HAS_AGG are compile-time so no v_cndmask selects land on the WMMA critical path.
template <bool RELU, bool HAS_AGG>
__global__ void k_gemm_node(const float* __restrict__ A, const float* __restrict__ W,
                            float* __restrict__ H, float* __restrict__ AGG,
                            const float* __restrict__ dinv, const float* __restrict__ bias,
                            int n) {
  const int m16  = blockIdx.x * 16;
  const int wave = threadIdx.x >> 5;   // 0..7 -> 16-col tile
  const int lane = threadIdx.x & 31;
  const int half = lane >> 4;          // 0: K pair {0,1}, 1: K pair {2,3}
  const int lr   = lane & 15;
  const int n16  = wave * 16;

  v8f acc = {0.f, 0.f, 0.f, 0.f, 0.f, 0.f, 0.f, 0.f};

  int rowA = m16 + lr; if (rowA >= n) rowA = n - 1;      // clamp (n is a multiple of 16 here)
  const float* __restrict__ arow = A + (size_t)rowA * HDIM + 2 * half;
  const float* __restrict__ wcol = W + (size_t)(2 * half) * HDIM + n16 + lr;

  #pragma unroll
  for (int k0 = 0; k0 < HDIM; k0 += 4) {
    float a0 = arow[k0];
    float a1 = arow[k0 + 1];
    if (RELU) { a0 = fmaxf(a0, 0.f); a1 = fmaxf(a1, 0.f); }
    v2f a; a.x = a0; a.y = a1;
    v2f b;
    b.x = wcol[(size_t)k0 * HDIM];
    b.y = wcol[(size_t)k0 * HDIM + HDIM];
    acc = __builtin_amdgcn_wmma_f32_16x16x4_f32(false, a, false, b, (short)0, acc, false, false);
  }

  #pragma unroll
  for (int r = 0; r < 8; ++r) {
    int row = m16 + r + 8 * half;     // C/D layout: M = r + 8*half
    int col = n16 + lr;               // N = lane&15
    if (row < n) {
      float v = acc[r];
      size_t idx = (size_t)row * HDIM + col;
      H[idx] = v;
      if (HAS_AGG) {
        float di = dinv[row];
        AGG[idx] = v * di * di + bias[col];   // self-loop term + bias
      }
    }
  }
}

// ---------------- edge scatter: agg[dst] += h[src] * dinv[src]*dinv[dst]
// 128-thread (4-wave) groups: e is wave-uniform, channel c varies across lanes (coalesced rows).
__global__ void k_scatter(const float* __restrict__ h, float* __restrict__ agg,
                          const int* __restrict__ src, const int* __restrict__ dst,
                          const float* __restrict__ dinv, int E) {
  int c    = threadIdx.x & (HDIM - 1);
  int grp  = (blockIdx.x * blockDim.x + threadIdx.x) >> 7;
  int ngrp = (gridDim.x * blockDim.x) >> 7;
  for (int e = grp; e < E; e += ngrp) {
    int s = src[e], d = dst[e];
    float coeff = dinv[s] * dinv[d];
    atomicAdd(&agg[(size_t)d * HDIM + c], h[(size_t)s * HDIM + c] * coeff);
  }
}

// ---------------- edge MLP + mean-reduce:
// gsum[c] += sum_e relu(P1[src[e]][c] + P2[dst[e]][c] + be[c] + ea[e,:]@WeE[:,c])
__global__ void k_edge_mlp(const float* __restrict__ P1, const float* __restrict__ P2,
                           const float* __restrict__ ea /*[E,7]*/,
                           const float* __restrict__ WeE /*[7,128]*/,
                           const float* __restrict__ be,
                           const int* __restrict__ src, const int* __restrict__ dst,
                           float* __restrict__ gsum, int E) {
  int c    = threadIdx.x & (HDIM - 1);
  int grp  = (blockIdx.x * blockDim.x + threadIdx.x) >> 7;
  int ngrp = (gridDim.x * blockDim.x) >> 7;
  float bec = be[c];
  float w0 = WeE[0 * HDIM + c], w1 = WeE[1 * HDIM + c], w2 = WeE[2 * HDIM + c];
  float w3 = WeE[3 * HDIM + c], w4 = WeE[4 * HDIM + c], w5 = WeE[5 * HDIM + c];
  float w6 = WeE[6 * HDIM + c];
  float local = 0.f;
  for (int e = grp; e < E; e += ngrp) {
    int s = src[e], d = dst[e];
    const float* a = ea + (size_t)e * 7;
    float v = P1[(size_t)s * HDIM + c] + P2[(size_t)d * HDIM + c] + bec;
    v = fmaf(a[0], w0, v); v = fmaf(a[1], w1, v); v = fmaf(a[2], w2, v);
    v = fmaf(a[3], w3, v); v = fmaf(a[4], w4, v); v = fmaf(a[5], w5, v);
    v = fmaf(a[6], w6, v);
    local += fmaxf(v, 0.f);
  }
  atomicAdd(&gsum[c], local);
}

__global__ void k_zero(float* p, int n) {
  int i = blockIdx.x * blockDim.x + threadIdx.x;
  if (i < n) p[i] = 0.f;
}

// out[0] = mean @ Wg + bg  (O == 1)
__global__ void k_finalize(const float* __restrict__ gsum, const float* __restrict__ Wg,
                           const float* __restrict__ bg, float* __restrict__ out, int E) {
  __shared__ float s[HDIM];
  int c = threadIdx.x;
  s[c] = gsum[c] * (1.0f / (float)E) * Wg[c];
  __syncthreads();
  #pragma unroll
  for (int off = HDIM / 2; off > 0; off >>= 1) {
    if (c < off) s[c] += s[c + off];
    __syncthreads();
  }
  if (c == 0) out[0] = s[0] + bg[0];
}

extern "C" void kernel_launch(void* const* d_in, const int* in_sizes, int n_in,
                              void* d_out, int out_size, void* d_ws, size_t ws_size,
                              hipStream_t stream) {
  const float* x  = (const float*)d_in[0];
  const int*   ei = (const int*)d_in[1];
  const float* ea = (const float*)d_in[2];
  const float* W1 = (const float*)d_in[3];  const float* b1 = (const float*)d_in[4];
  const float* W2 = (const float*)d_in[5];  const float* b2 = (const float*)d_in[6];
  const float* W3 = (const float*)d_in[7];  const float* b3 = (const float*)d_in[8];
  const float* We = (const float*)d_in[9];  const float* be = (const float*)d_in[10];
  const float* Wg = (const float*)d_in[11]; const float* bg = (const float*)d_in[12];
  float* out = (float*)d_out;

  const int N = in_sizes[0] / HDIM;
  const int E = in_sizes[1] / 2;
  const int T = in_sizes[2] / (E * 7);
  const int* src  = ei;
  const int* dstp = ei + E;

  float* ws   = (float*)d_ws;
  float* dinv = ws;  ws += N;
  float* h    = ws;  ws += (size_t)N * HDIM;   // h, also reused as P1
  float* X0   = ws;  ws += (size_t)N * HDIM;
  float* X1   = ws;  ws += (size_t)N * HDIM;
  float* gsum = ws;  ws += HDIM;

  // dinv = rsqrt(deg+1), depends only on dst -> compute once
  k_deg_init<<<(N + 255) / 256, 256, 0, stream>>>(dinv, N);
  k_deg_count<<<1024, 256, 0, stream>>>(dstp, dinv, E);
  k_deg_fin<<<(N + 255) / 256, 256, 0, stream>>>(dinv, N);

  const int mtiles = (N + 15) / 16;
  const float* Ws[3] = {W1, W2, W3};
  const float* bs[3] = {b1, b2, b3};

  const float* xin = x;   // current layer input (pre-activation; relu applied on load)
  float* ping = X0;
  float* pong = X1;

  for (int t = 0; t < T; ++t) {
    for (int l = 0; l < 3; ++l) {
      // h = relu?(xin) @ W ; ping initialized with self-loop + bias
      if (t == 0 && l == 0)
        k_gemm_node<false, true><<<mtiles, 256, 0, stream>>>(xin, Ws[l], h, ping, dinv, bs[l], N);
      else
        k_gemm_node<true, true><<<mtiles, 256, 0, stream>>>(xin, Ws[l], h, ping, dinv, bs[l], N);
      k_scatter<<<4096, 256, 0, stream>>>(h, ping, src, dstp, dinv, E);
      xin = ping;                         // pre-activation of this layer
      float* tmp = ping; ping = pong; pong = tmp;
    }
    // Edge phase: P1 = relu(x3)@We[0:128], P2 = relu(x3)@We[128:256]
    float* P1 = h;       // h is free after last scatter
    float* P2 = ping;    // free ping-pong buffer
    k_gemm_node<true, false><<<mtiles, 256, 0, stream>>>(xin, We, P1, nullptr, nullptr, nullptr, N);
    k_gemm_node<true, false><<<mtiles, 256, 0, stream>>>(xin, We + 128 * HDIM, P2, nullptr, nullptr, nullptr, N);
    k_zero<<<1, HDIM, 0, stream>>>(gsum, HDIM);
    k_edge_mlp<<<4096, 256, 0, stream>>>(P1, P2, ea + (size_t)t * E * 7,
                                         We + 256 * HDIM, be, src, dstp, gsum, E);
    k_finalize<<<1, HDIM, 0, stream>>>(gsum, Wg, bg, out + t, E);
  }
}